// Lukasiewicz_56564719288625
// MI455X (gfx1250) — compile-verified
//
#include <hip/hip_runtime.h>
#include <hip/hip_bf16.h>

// Problem sizes (fixed by the reference).
#define N_ROWS 2048
#define IN_F   512
#define OUT_F  512

// Tiling.
#define BN 64          // rows of x per block
#define BO 64          // rows of a (output cols) per block
#define BK 64          // K chunk
#define NTHREADS 256   // 8 wave32s
#define TILE_F (BN * BK)   // floats per tile buffer (4096)

typedef float v4f __attribute__((ext_vector_type(4)));

// ---- CDNA5 async global->LDS path (ASYNCcnt), with sync fallback ----
#if defined(__has_builtin)
#  if __has_builtin(__builtin_amdgcn_global_load_async_to_lds_b128)
#    define HAVE_ASYNC_LDS 1
#  endif
#endif
#ifndef HAVE_ASYNC_LDS
#  define HAVE_ASYNC_LDS 0
#endif

#if HAVE_ASYNC_LDS
typedef int v4i __attribute__((vector_size(4 * sizeof(int))));
typedef __attribute__((address_space(1))) v4i as1_v4i;
typedef __attribute__((address_space(3))) v4i as3_v4i;

__device__ __forceinline__ void async_cp16(const float* g, float* l) {
  // 16B per lane, global -> LDS, tracked by ASYNCcnt (GLOBAL_LOAD_ASYNC_TO_LDS_B128)
  __builtin_amdgcn_global_load_async_to_lds_b128((as1_v4i*)g, (as3_v4i*)l, 0, 0);
}
__device__ __forceinline__ void wait_async0() {
#if __has_builtin(__builtin_amdgcn_s_wait_asynccnt)
  __builtin_amdgcn_s_wait_asynccnt(0);
#else
  asm volatile("s_wait_asynccnt 0" ::: "memory");
#endif
}
#endif

// Compute one 64x64x64 tile stage from LDS.
// xsb: x tile [row(0..63)][k(0..63)], row stride 64 floats (reads are lane-broadcast).
// asb: a tile, 16B-block XOR swizzled: logical (row, kb) lives at block kb ^ (row & 15)
//      -> conflict-free b128 reads across the 16 lanes of a half-wave.
// NOTE: unroll 4 (not full) keeps ~2 k-steps of xv/av live -> no spills, <256 VGPRs.
__device__ __forceinline__ void tile_compute(const float* xsb, const float* asb,
                                             int tx, int ty, float acc[4][4]) {
#pragma unroll 4
  for (int kb = 0; kb < BK / 4; ++kb) {
    v4f xv[4], av[4];
#pragma unroll
    for (int i = 0; i < 4; ++i)
      xv[i] = *(const v4f*)(xsb + (ty + 16 * i) * BK + kb * 4);
#pragma unroll
    for (int j = 0; j < 4; ++j)
      av[j] = *(const v4f*)(asb + (tx + 16 * j) * BK + ((kb ^ tx) & 15) * 4);
#pragma unroll
    for (int i = 0; i < 4; ++i) {
#pragma unroll
      for (int j = 0; j < 4; ++j) {
        v4f s = xv[i] + av[j];                        // 2x v_pk_add_f32
        float m = fmaxf(fmaxf(s.x, s.y), s.z);        // v_max3_f32
        acc[i][j] = fmaxf(fmaxf(m, s.w), acc[i][j]);  // v_max3_f32
      }
    }
  }
}

__global__ __launch_bounds__(NTHREADS)
void lukasiewicz_maxplus_kernel(const float* __restrict__ X,
                                const float* __restrict__ A,
                                const float* __restrict__ B,
                                float* __restrict__ Y) {
  extern __shared__ float smem[];
  float* xs  = smem;               // 2 buffers of 64x64 floats (16 KB each)
  float* as_ = smem + 2 * TILE_F;  // 2 buffers of 64x64 floats (XOR-swizzled)

  const int tid = threadIdx.x;
  const int tx  = tid & 15;   // output-col lane: o = o0 + tx + 16j
  const int ty  = tid >> 4;   // row lane:        n = n0 + ty + 16i
  const int o0  = blockIdx.x * BO;
  const int n0  = blockIdx.y * BN;

  // Loader mapping: each thread moves 8 x 16B (4 for x, 4 for a) per K-stage.
  const int kb = tid & 15;    // 16B block index along K
  const int r0 = tid >> 4;    // base row, rows r0 + 16p

  float acc[4][4];
#pragma unroll
  for (int i = 0; i < 4; ++i)
#pragma unroll
    for (int j = 0; j < 4; ++j)
      acc[i][j] = -3.402823466e+38f;   // values are finite sums in [0,2]

  const int KT = IN_F / BK;   // 8 stages

  // Per-thread base pointers / LDS offsets, advanced by BK per stage.
  const float* gx[4];
  const float* ga[4];
  int lxo[4], lao[4];
#pragma unroll
  for (int p = 0; p < 4; ++p) {
    const int r = r0 + 16 * p;
    gx[p]  = X + (size_t)(n0 + r) * IN_F + kb * 4;
    ga[p]  = A + (size_t)(o0 + r) * IN_F + kb * 4;
    lxo[p] = r * BK + kb * 4;
    lao[p] = r * BK + (kb ^ r0) * 4;
  }

#if HAVE_ASYNC_LDS
  // -------- double-buffered async pipeline --------
  {
    // prefetch stage 0 into buffer 0
#pragma unroll
    for (int p = 0; p < 4; ++p) {
      async_cp16(gx[p], xs + lxo[p]);
      async_cp16(ga[p], as_ + lao[p]);
    }
    wait_async0();
    __syncthreads();

    int buf = 0;
#pragma unroll 1
    for (int t = 0; t < KT; ++t) {
      if (t + 1 < KT) {
        float* xsb = xs  + (buf ^ 1) * TILE_F;
        float* asb = as_ + (buf ^ 1) * TILE_F;
#pragma unroll
        for (int p = 0; p < 4; ++p) {
          gx[p] += BK;
          ga[p] += BK;
          async_cp16(gx[p], xsb + lxo[p]);
          async_cp16(ga[p], asb + lao[p]);
        }
      }
      tile_compute(xs + buf * TILE_F, as_ + buf * TILE_F, tx, ty, acc);
      wait_async0();
      __syncthreads();
      buf ^= 1;
    }
  }
#else
  // -------- synchronous fallback (single buffer) --------
#pragma unroll 1
  for (int t = 0; t < KT; ++t) {
    __syncthreads();
#pragma unroll
    for (int p = 0; p < 4; ++p) {
      float4 vx = *(const float4*)(gx[p]);
      float4 va = *(const float4*)(ga[p]);
      gx[p] += BK;
      ga[p] += BK;
      *(float4*)(xs  + lxo[p]) = vx;
      *(float4*)(as_ + lao[p]) = va;
    }
    __syncthreads();
    tile_compute(xs, as_, tx, ty, acc);
  }
#endif

  // Epilogue: y = max(max_k - 1, 0, b[o])   (the -1 hoisted out of the K loop)
  float bv[4];
#pragma unroll
  for (int j = 0; j < 4; ++j)
    bv[j] = B[o0 + tx + 16 * j];

#pragma unroll
  for (int i = 0; i < 4; ++i) {
    const int n = n0 + ty + 16 * i;
#pragma unroll
    for (int j = 0; j < 4; ++j) {
      const int o = o0 + tx + 16 * j;
      float r = fmaxf(fmaxf(acc[i][j] - 1.0f, 0.0f), bv[j]);  // v_max3_f32
      Y[(size_t)n * OUT_F + o] = r;
    }
  }
}

extern "C" void kernel_launch(void* const* d_in, const int* in_sizes, int n_in,
                              void* d_out, int out_size, void* d_ws, size_t ws_size,
                              hipStream_t stream) {
  const float* x = (const float*)d_in[0];   // [2048, 512]
  const float* a = (const float*)d_in[1];   // [512, 512]
  const float* b = (const float*)d_in[2];   // [512]
  float* y = (float*)d_out;                 // [2048, 512]

  dim3 grid(OUT_F / BO, N_ROWS / BN);       // (8, 32) = 256 blocks
  dim3 block(NTHREADS);
  size_t shmem = (size_t)4 * TILE_F * sizeof(float);  // 64 KiB: 2x double-buffered tiles

  hipLaunchKernelGGL(lukasiewicz_maxplus_kernel, grid, block, shmem, stream,
                     x, a, b, y);
}